// GCNet_34488587387425
// MI455X (gfx1250) — compile-verified
//
#include <hip/hip_runtime.h>

#define NN 100000
#define EE 1600000
#define ETOT (EE + NN)     // edges + self loops
#define VIN 64
#define CC 64
#define HH 4
#define HC 256             // HH * CC
#define NETD 128
#define GG 128
#define SLOPE 0.2f

typedef __attribute__((ext_vector_type(2))) float v2f;
typedef __attribute__((ext_vector_type(8))) float v8f;

// ---------------- utility kernels ----------------

__global__ void k_fill(float* __restrict__ p, float v, long n) {
    long i = (long)blockIdx.x * blockDim.x + threadIdx.x;
    if (i < n) p[i] = v;
}

__global__ void k_deg(const int* __restrict__ ei, float* __restrict__ deg) {
    long e = (long)blockIdx.x * blockDim.x + threadIdx.x;
    if (e < EE) atomicAdd(&deg[ei[EE + e]], 1.0f);
}

__global__ void k_dinv(float* __restrict__ deg) {
    long i = (long)blockIdx.x * blockDim.x + threadIdx.x;
    if (i < NN) deg[i] = rsqrtf(deg[i] + 1.0f);   // +1 for self loop
}

__global__ void k_bias_relu(float* __restrict__ p, const float* __restrict__ b,
                            long n, int cmask) {
    long i = (long)blockIdx.x * blockDim.x + threadIdx.x;
    if (i < n) p[i] = fmaxf(p[i] + b[i & cmask], 0.0f);
}

// ---------------- WMMA fp32 GEMM: C = [relu](A[M,K] @ B[K,N] (+bias)) ----------------
// Block = 256 threads (8 waves). Whole B (K*Nc <= 16384 floats = 64KB) is staged
// into LDS once per block; each wave owns 16 rows of A and sweeps N in chunks of
// 4 tiles (64 cols), reusing each A fragment across 4 WMMAs.
// V_WMMA_F32_16X16X4_F32 layouts (ISA 7.12.2):
//   A vgpr0/1 = K,K+1 (lanes 0-15) / K+2,K+3 (lanes 16-31), M = lane&15
//   B symmetric; C/D vgpr r = rows M=r (lanes 0-15) and M=r+8 (lanes 16-31).

__global__ void k_wmma_gemm(const float* __restrict__ A, const float* __restrict__ B,
                            const float* __restrict__ bias, float* __restrict__ C,
                            int M, int K, int Nc, int relu) {
    __shared__ float Bs[16384];                 // 64 KB of 320 KB/WGP

    const int tid = threadIdx.x;
    const int total = K * Nc;
    for (int i = tid; i < total; i += 256)      // cooperative, coalesced B stage
        Bs[i] = B[i];
    __syncthreads();

    const int wave = tid >> 5;                  // 0..7
    const int lane = tid & 31;
    const int half = lane >> 4;                 // 0: K,K+1 ; 1: K+2,K+3
    const int l15  = lane & 15;
    const int tm   = (blockIdx.x * 8 + wave) << 4;
    if (tm >= M) return;                        // wave-uniform: EXEC all-ones below

    const int tilesN = Nc >> 4;
    const float* arow = A + (size_t)(tm + l15) * K;

    for (int tc = 0; tc < tilesN; tc += 4) {    // 4-tile (64-col) N chunks
        v8f acc[4] = {};
        for (int k = 0; k < K; k += 4) {
            const int ka = k + 2 * half;
            v2f a = *(const v2f*)(arow + ka);   // one global_load_b64 per lane
            __builtin_prefetch(arow + ka + 16, 0, 1);
#pragma unroll
            for (int j = 0; j < 4; ++j) {
                const int tn = (tc + j) << 4;
                v2f b;
                b.x = Bs[ka * Nc + tn + l15];   // ds_load from staged B
                b.y = Bs[(ka + 1) * Nc + tn + l15];
                acc[j] = __builtin_amdgcn_wmma_f32_16x16x4_f32(
                    /*neg_a=*/false, a, /*neg_b=*/false, b,
                    /*c_mod=*/(short)0, acc[j],
                    /*reuse_a=*/false, /*reuse_b=*/false);
            }
        }
#pragma unroll
        for (int j = 0; j < 4; ++j) {
            const int tn  = (tc + j) << 4;
            const int col = tn + l15;
            float bb = bias ? bias[col] : 0.0f;
#pragma unroll
            for (int r = 0; r < 8; ++r) {
                const int row = tm + r + 8 * half;
                float v = acc[j][r] + bb;
                if (relu) v = fmaxf(v, 0.0f);
                C[(size_t)row * Nc + col] = v;
            }
        }
    }
}

// ---------------- GCN aggregation: out[dst] += xw[src] * dinv[src]*dinv[dst] ----------------
// float4 per thread: one b128 gather + 4 atomics.

__global__ void k_gcn_agg(const int* __restrict__ ei, const float* __restrict__ xw,
                          const float* __restrict__ dinv, float* __restrict__ out) {
    long idx = (long)blockIdx.x * blockDim.x + threadIdx.x;
    if (idx >= (long)ETOT * (CC / 4)) return;
    int  c4 = (int)(idx & (CC / 4 - 1)) << 2;
    long e  = idx >> 4;                         // CC/4 = 16
    int s, d;
    if (e < EE) { s = ei[e]; d = ei[EE + e]; } else { s = d = (int)(e - EE); }
    float norm = dinv[s] * dinv[d];
    float4 v = *(const float4*)(xw + (size_t)s * CC + c4);
    float* o = out + (size_t)d * CC + c4;
    atomicAdd(o + 0, v.x * norm);
    atomicAdd(o + 1, v.y * norm);
    atomicAdd(o + 2, v.z * norm);
    atomicAdd(o + 3, v.w * norm);
}

// ---------------- GAT ----------------

__device__ __forceinline__ float lrelu(float v) { return v > 0.0f ? v : SLOPE * v; }

// order-preserving float <-> uint for atomicMax
__device__ __forceinline__ unsigned f2ord(float f) {
    unsigned b = __float_as_uint(f);
    return (b & 0x80000000u) ? ~b : (b | 0x80000000u);
}
__device__ __forceinline__ float ord2f(unsigned u) {
    unsigned b = (u & 0x80000000u) ? (u ^ 0x80000000u) : ~u;
    return __uint_as_float(b);
}

__global__ void k_att(const float* __restrict__ xh, const float* __restrict__ att_s,
                      const float* __restrict__ att_d, float* __restrict__ asrc,
                      float* __restrict__ adst) {
    long i = (long)blockIdx.x * blockDim.x + threadIdx.x;   // i = n*HH + h
    if (i >= (long)NN * HH) return;
    int n = (int)(i >> 2), h = (int)(i & 3);
    const float* xr = xh + (size_t)n * HC + h * CC;
    float ss = 0.0f, sd = 0.0f;
    for (int c = 0; c < CC; c += 4) {
        float4 v = *(const float4*)(xr + c);
        float4 ws = *(const float4*)(att_s + h * CC + c);
        float4 wd = *(const float4*)(att_d + h * CC + c);
        ss += v.x * ws.x + v.y * ws.y + v.z * ws.z + v.w * ws.w;
        sd += v.x * wd.x + v.y * wd.y + v.z * wd.z + v.w * wd.w;
    }
    asrc[i] = ss;
    adst[i] = sd;
}

__global__ void k_edge_max(const int* __restrict__ ei, const float* __restrict__ asrc,
                           const float* __restrict__ adst, unsigned* __restrict__ emax) {
    long idx = (long)blockIdx.x * blockDim.x + threadIdx.x;
    if (idx >= (long)ETOT * HH) return;
    long e = idx >> 2; int h = (int)(idx & 3);
    int s, d;
    if (e < EE) { s = ei[e]; d = ei[EE + e]; } else { s = d = (int)(e - EE); }
    float v = lrelu(asrc[(size_t)s * HH + h] + adst[(size_t)d * HH + h]);
    atomicMax(&emax[(size_t)d * HH + h], f2ord(v));
}

__global__ void k_decode_max(unsigned* __restrict__ emax) {
    long i = (long)blockIdx.x * blockDim.x + threadIdx.x;
    if (i < (long)NN * HH) ((float*)emax)[i] = ord2f(emax[i]);
}

__global__ void k_edge_exp(const int* __restrict__ ei, const float* __restrict__ asrc,
                           const float* __restrict__ adst, const float* __restrict__ emax,
                           float* __restrict__ ebuf, float* __restrict__ denom) {
    long idx = (long)blockIdx.x * blockDim.x + threadIdx.x;
    if (idx >= (long)ETOT * HH) return;
    long e = idx >> 2; int h = (int)(idx & 3);
    int s, d;
    if (e < EE) { s = ei[e]; d = ei[EE + e]; } else { s = d = (int)(e - EE); }
    float v  = lrelu(asrc[(size_t)s * HH + h] + adst[(size_t)d * HH + h]);
    float ee = __expf(v - emax[(size_t)d * HH + h]);
    ebuf[idx] = ee;
    atomicAdd(&denom[(size_t)d * HH + h], ee);
}

__global__ void k_gat_agg(const int* __restrict__ ei, const float* __restrict__ xh,
                          const float* __restrict__ ebuf, const float* __restrict__ denom,
                          float* __restrict__ out) {
    long idx = (long)blockIdx.x * blockDim.x + threadIdx.x;
    if (idx >= (long)ETOT * (HC / 4)) return;
    int  r4 = (int)(idx & (HC / 4 - 1)) << 2;   // r4 = h*CC + c, c multiple of 4
    long e  = idx >> 6;                         // HC/4 = 64
    int s, d;
    if (e < EE) { s = ei[e]; d = ei[EE + e]; } else { s = d = (int)(e - EE); }
    int h = r4 >> 6;
    float alpha = ebuf[e * HH + h] / denom[(size_t)d * HH + h];
    float4 v = *(const float4*)(xh + (size_t)s * HC + r4);
    float* o = out + (size_t)d * HC + r4;
    atomicAdd(o + 0, v.x * alpha);
    atomicAdd(o + 1, v.y * alpha);
    atomicAdd(o + 2, v.z * alpha);
    atomicAdd(o + 3, v.w * alpha);
}

// ---------------- pooling + MLP tail ----------------

__global__ void k_pool(const int* __restrict__ batch, const float* __restrict__ h,
                       float* __restrict__ pooled) {
    long idx = (long)blockIdx.x * blockDim.x + threadIdx.x;
    if (idx >= (long)NN * (CC / 4)) return;
    int c4 = (int)(idx & (CC / 4 - 1)) << 2;
    int n  = (int)(idx >> 4);
    float4 v = *(const float4*)(h + (size_t)n * CC + c4);
    float* o = pooled + (size_t)batch[n] * CC + c4;
    atomicAdd(o + 0, v.x);
    atomicAdd(o + 1, v.y);
    atomicAdd(o + 2, v.z);
    atomicAdd(o + 3, v.w);
}

__global__ void k_final(const float* __restrict__ m2, const float* __restrict__ M3,
                        const float* __restrict__ bm3, float* __restrict__ out) {
    int g = blockIdx.x * blockDim.x + threadIdx.x;
    if (g >= GG) return;
    float s = bm3[0];
    for (int j = 0; j < NETD; ++j) s += m2[g * NETD + j] * M3[j];
    out[g] = s;
}

// ---------------- launch ----------------

extern "C" void kernel_launch(void* const* d_in, const int* in_sizes, int n_in,
                              void* d_out, int out_size, void* d_ws, size_t ws_size,
                              hipStream_t stream) {
    (void)in_sizes; (void)n_in; (void)out_size; (void)ws_size;

    const float* x     = (const float*)d_in[0];
    const int*   ei    = (const int*)d_in[1];
    const int*   batch = (const int*)d_in[2];
    const float* W1    = (const float*)d_in[3];
    const float* b1    = (const float*)d_in[4];
    const float* Wg    = (const float*)d_in[5];
    const float* atts  = (const float*)d_in[6];
    const float* attd  = (const float*)d_in[7];
    const float* bg    = (const float*)d_in[8];
    const float* W2    = (const float*)d_in[9];
    const float* b2    = (const float*)d_in[10];
    const float* M1    = (const float*)d_in[11];
    const float* bm1   = (const float*)d_in[12];
    const float* M2    = (const float*)d_in[13];
    const float* bm2   = (const float*)d_in[14];
    const float* M3    = (const float*)d_in[15];
    const float* bm3   = (const float*)d_in[16];
    float* out = (float*)d_out;

    float* ws = (float*)d_ws;
    size_t off = 0;
    auto alloc = [&](size_t n) { float* p = ws + off; off += n; return p; };
    float*    xh     = alloc((size_t)NN * HC);     // GAT features (L2-resident, 102MB)
    float*    gat    = alloc((size_t)NN * HC);     // GAT output accumulator
    float*    h1     = alloc((size_t)NN * CC);     // GCN1 out; reused as GCN2 out (h3)
    float*    xw     = alloc((size_t)NN * CC);     // xw1, reused as xw2
    float*    dinv   = alloc((size_t)NN);
    float*    asrc   = alloc((size_t)NN * HH);
    float*    adst   = alloc((size_t)NN * HH);
    unsigned* emax   = (unsigned*)alloc((size_t)NN * HH);
    float*    denom  = alloc((size_t)NN * HH);
    float*    ebuf   = alloc((size_t)ETOT * HH);
    float*    pooled = alloc((size_t)GG * CC);
    float*    m1     = alloc((size_t)GG * NETD);
    float*    m2     = alloc((size_t)GG * NETD);

    const int T = 256;
    auto nb = [](long n) { return (unsigned)((n + 255) / 256); };
    auto gemm_blocks = [](int M) {               // 8 row-tiles (128 rows) per block
        return (unsigned)((M / 16 + 7) / 8);
    };

    // degree / normalization
    k_fill<<<nb(NN), T, 0, stream>>>(dinv, 0.0f, (long)NN);
    k_deg <<<nb(EE), T, 0, stream>>>(ei, dinv);
    k_dinv<<<nb(NN), T, 0, stream>>>(dinv);

    // GCN1: h1 = relu(Dn (A+I) Dn (x @ W1) + b1)
    k_wmma_gemm<<<gemm_blocks(NN), T, 0, stream>>>(x, W1, nullptr, xw, NN, VIN, CC, 0);
    k_fill     <<<nb((long)NN * CC), T, 0, stream>>>(h1, 0.0f, (long)NN * CC);
    k_gcn_agg  <<<nb((long)ETOT * (CC / 4)), T, 0, stream>>>(ei, xw, dinv, h1);
    k_bias_relu<<<nb((long)NN * CC), T, 0, stream>>>(h1, b1, (long)NN * CC, CC - 1);

    // GAT: xh = h1 @ Wg; per-head softmax attention; gat = relu(agg + bg)
    k_wmma_gemm<<<gemm_blocks(NN), T, 0, stream>>>(h1, Wg, nullptr, xh, NN, CC, HC, 0);
    k_att      <<<nb((long)NN * HH), T, 0, stream>>>(xh, atts, attd, asrc, adst);
    k_fill     <<<nb((long)NN * HH), T, 0, stream>>>((float*)emax, 0.0f, (long)NN * HH); // ord-min
    k_fill     <<<nb((long)NN * HH), T, 0, stream>>>(denom, 0.0f, (long)NN * HH);
    k_fill     <<<nb((long)NN * HC), T, 0, stream>>>(gat, 0.0f, (long)NN * HC);
    k_edge_max <<<nb((long)ETOT * HH), T, 0, stream>>>(ei, asrc, adst, emax);
    k_decode_max<<<nb((long)NN * HH), T, 0, stream>>>(emax);
    k_edge_exp <<<nb((long)ETOT * HH), T, 0, stream>>>(ei, asrc, adst, (const float*)emax, ebuf, denom);
    k_gat_agg  <<<nb((long)ETOT * (HC / 4)), T, 0, stream>>>(ei, xh, ebuf, denom, gat);
    k_bias_relu<<<nb((long)NN * HC), T, 0, stream>>>(gat, bg, (long)NN * HC, HC - 1);

    // GCN2: h3 = relu(Dn (A+I) Dn (gat @ W2) + b2)   (h3 reuses h1 storage)
    k_wmma_gemm<<<gemm_blocks(NN), T, 0, stream>>>(gat, W2, nullptr, xw, NN, HC, CC, 0);
    float* h3 = h1;
    k_fill     <<<nb((long)NN * CC), T, 0, stream>>>(h3, 0.0f, (long)NN * CC);
    k_gcn_agg  <<<nb((long)ETOT * (CC / 4)), T, 0, stream>>>(ei, xw, dinv, h3);
    k_bias_relu<<<nb((long)NN * CC), T, 0, stream>>>(h3, b2, (long)NN * CC, CC - 1);

    // global_add_pool -> [G, C]
    k_fill<<<nb((long)GG * CC), T, 0, stream>>>(pooled, 0.0f, (long)GG * CC);
    k_pool<<<nb((long)NN * (CC / 4)), T, 0, stream>>>(batch, h3, pooled);

    // MLP head (bias+relu fused into WMMA GEMM epilogue)
    k_wmma_gemm<<<gemm_blocks(GG), T, 0, stream>>>(pooled, M1, bm1, m1, GG, CC,   NETD, 1);
    k_wmma_gemm<<<gemm_blocks(GG), T, 0, stream>>>(m1,     M2, bm2, m2, GG, NETD, NETD, 1);
    k_final<<<1, GG, 0, stream>>>(m2, M3, bm3, out);
}